// LQR_83434034692542
// MI455X (gfx1250) — compile-verified
//
#include <hip/hip_runtime.h>
#include <hip/hip_bf16.h>

// ---------------------------------------------------------------------------
// LQR on MI455X (gfx1250): persistent single-WGP Riccati recursion using
// V_WMMA_F32_16X16X4_F32 for all dense products, all state in 320KB LDS,
// followed by a sequential rollout kernel.
// ---------------------------------------------------------------------------

typedef float v2f __attribute__((ext_vector_type(2)));
typedef float v8f __attribute__((ext_vector_type(8)));

#define LQR_T      2048
#define LQR_NX     128
#define LQR_NU     64
#define LQR_ND     192
#define RIC_NT     512   // 16 waves
#define RIC_NW     16

// --- WMMA f32 16x16x4 wrapper ----------------------------------------------
__device__ __forceinline__ v8f wmma_f32(v2f a, v2f b, v8f c) {
  // 8 args: (neg_a, A, neg_b, B, c_mod, C, reuse_a, reuse_b)
  return __builtin_amdgcn_wmma_f32_16x16x4_f32(
      false, a, false, b, (short)0, c, false, false);
}

// A fragment (16x4) from row-major src: A[m,k] = p[(row0+m)*ld + k0+k]
// layout: lane = m + 16*(k>=2); vgpr = k&1
__device__ __forceinline__ v2f load_a_rm(const float* p, int ld, int row0,
                                         int k0, int lane) {
  int m  = lane & 15;
  int kb = (lane >> 4) * 2;
  const float* q = p + (row0 + m) * ld + k0 + kb;
  v2f a; a.x = q[0]; a.y = q[1];
  return a;
}

// A fragment (16x4) transposed: A[m,k] = p[(k0+k)*ld + col0+m]
__device__ __forceinline__ v2f load_a_tr(const float* p, int ld, int col0,
                                         int k0, int lane) {
  int m  = lane & 15;
  int kb = (lane >> 4) * 2;
  const float* q = p + (k0 + kb) * ld + col0 + m;
  v2f a; a.x = q[0]; a.y = q[ld];
  return a;
}

// B fragment (4x16) from row-major src: B[k,n] = p[(k0+k)*ld + col0+n]
// layout: lane = n + 16*(k>=2); vgpr = k&1
__device__ __forceinline__ v2f load_b_rm(const float* p, int ld, int k0,
                                         int col0, int lane) {
  int n  = lane & 15;
  int kb = (lane >> 4) * 2;
  const float* q = p + (k0 + kb) * ld + col0 + n;
  v2f b; b.x = q[0]; b.y = q[ld];
  return b;
}

// C/D fragment (16x16 f32): elem(m,n) lane = n + 16*(m>=8); vgpr = m&7
__device__ __forceinline__ v8f load_c_rm(const float* p, int ld, int row0,
                                         int col0, int lane) {
  int n  = lane & 15;
  int mh = (lane >> 4) * 8;
  v8f c;
#pragma unroll
  for (int i = 0; i < 8; ++i) c[i] = p[(row0 + mh + i) * ld + col0 + n];
  return c;
}

__device__ __forceinline__ void store_c_rm(float* p, int ld, int row0,
                                           int col0, v8f c, int lane) {
  int n  = lane & 15;
  int mh = (lane >> 4) * 8;
#pragma unroll
  for (int i = 0; i < 8; ++i) p[(row0 + mh + i) * ld + col0 + n] = c[i];
}

__device__ __forceinline__ void store_c_rm_neg(float* p, int ld, int row0,
                                               int col0, v8f c, int lane) {
  int n  = lane & 15;
  int mh = (lane >> 4) * 8;
#pragma unroll
  for (int i = 0; i < 8; ++i) p[(row0 + mh + i) * ld + col0 + n] = -c[i];
}

// ---------------------------------------------------------------------------
// Kernel 1: backward Riccati recursion. One workgroup, 16 waves, state in LDS.
// LDS float layout (70208 floats = 280832 B <= 320KB):
//   V    @ 0      (128x128)
//   M    @ 16384  (128x192)   reused after Q: Inv@16384, Work@20480, K@24576
//   Qxx  @ 40960  (128x128)
//   Qux  @ 57344  ( 64x128)
//   Quu  @ 65536  ( 64x 64)
//   v    @ 69632, w @ 69760, q @ 69888, kv @ 70080, fac @ 70144
// ---------------------------------------------------------------------------
__global__ void __launch_bounds__(RIC_NT)
lqr_riccati(const float* __restrict__ F, const float* __restrict__ fv,
            const float* __restrict__ C, const float* __restrict__ cv,
            float* __restrict__ Kout, float* __restrict__ kout) {
  extern __shared__ float lds[];
  float* V   = lds;
  float* M   = lds + 16384;
  float* Inv = lds + 16384;
  float* Wk  = lds + 20480;
  float* Km  = lds + 24576;
  float* Qxx = lds + 40960;
  float* Qux = lds + 57344;
  float* Quu = lds + 65536;
  float* vv  = lds + 69632;
  float* ww  = lds + 69760;
  float* qq  = lds + 69888;
  float* kv  = lds + 70080;
  float* fac = lds + 70144;

  const int tid  = threadIdx.x;
  const int lane = tid & 31;
  const int wave = tid >> 5;

  // V0 = C[0:128, 0:128], v0 = c[0:128]
  for (int i = tid; i < LQR_NX * LQR_NX; i += RIC_NT) {
    int r = i >> 7, cc = i & 127;
    V[i] = C[r * LQR_ND + cc];
  }
  for (int i = tid; i < LQR_NX; i += RIC_NT) vv[i] = cv[i];
  __syncthreads();

  for (int t = 0; t < LQR_T; ++t) {
    // ---- phase 1: w = V f + v  and  M = V @ F (WMMA, 96 tiles) ----
    for (int i = tid; i < LQR_NX; i += RIC_NT) {
      float s = vv[i];
      const float* Vr = V + i * LQR_NX;
      for (int k = 0; k < LQR_NX; ++k) s += Vr[k] * fv[k];
      ww[i] = s;
    }
    for (int tile = wave; tile < 96; tile += RIC_NW) {
      int r0 = (tile / 12) * 16, c0 = (tile % 12) * 16;
      v8f acc = {};
      for (int k0 = 0; k0 < LQR_NX; k0 += 4) {
        v2f a = load_a_rm(V, LQR_NX, r0, k0, lane);
        v2f b = load_b_rm(F, LQR_ND, k0, c0, lane);
        acc = wmma_f32(a, b, acc);
      }
      store_c_rm(M, LQR_ND, r0, c0, acc, lane);
    }
    __syncthreads();

    // ---- phase 2: q = c + F^T w  and  Q = C + F^T M (112 tiles) ----
    for (int i = tid; i < LQR_ND; i += RIC_NT) {
      float s = cv[i];
      for (int k = 0; k < LQR_NX; ++k) s += F[k * LQR_ND + i] * ww[k];
      qq[i] = s;
    }
    for (int tile = wave; tile < 112; tile += RIC_NW) {
      int r0, c0, dld, dr0, dc0;
      float* dst;
      if (tile < 64) {               // Qxx block
        r0 = (tile >> 3) * 16; c0 = (tile & 7) * 16;
        dst = Qxx; dld = LQR_NX; dr0 = r0; dc0 = c0;
      } else {                       // Qux / Quu block rows
        int u = tile - 64;
        int tr = u / 12, tc = u % 12;
        r0 = LQR_NX + tr * 16; c0 = tc * 16;
        if (tc < 8) { dst = Qux; dld = LQR_NX; dr0 = tr * 16; dc0 = c0; }
        else        { dst = Quu; dld = LQR_NU; dr0 = tr * 16; dc0 = (tc - 8) * 16; }
      }
      v8f acc = load_c_rm(C, LQR_ND, r0, c0, lane);
      for (int k0 = 0; k0 < LQR_NX; k0 += 4) {
        v2f a = load_a_tr(F, LQR_ND, r0, k0, lane);  // (F^T)[m,k] = F[k, r0+m]
        v2f b = load_b_rm(M, LQR_ND, k0, c0, lane);
        acc = wmma_f32(a, b, acc);
      }
      store_c_rm(dst, dld, dr0, dc0, acc, lane);
    }
    __syncthreads();

    // ---- phase 3: invert SPD Quu (pivot-free Gauss-Jordan in LDS) ----
    for (int i = tid; i < LQR_NU * LQR_NU; i += RIC_NT) {
      Wk[i]  = Quu[i];
      Inv[i] = ((i >> 6) == (i & 63)) ? 1.0f : 0.0f;
    }
    __syncthreads();
    for (int p = 0; p < LQR_NU; ++p) {
      if (tid < LQR_NU) fac[tid] = Wk[tid * LQR_NU + p];
      __syncthreads();
      float pinv = 1.0f / fac[p];
      for (int j = tid; j < LQR_NU; j += RIC_NT) {
        Wk[p * LQR_NU + j]  *= pinv;
        Inv[p * LQR_NU + j] *= pinv;
      }
      __syncthreads();
      for (int idx = tid; idx < LQR_NU * LQR_NU; idx += RIC_NT) {
        int r = idx >> 6, j = idx & 63;
        if (r != p) {
          float fr = fac[r];
          Wk[idx]  -= fr * Wk[p * LQR_NU + j];
          Inv[idx] -= fr * Inv[p * LQR_NU + j];
        }
      }
      __syncthreads();
    }

    // ---- phase 4: K = -Inv @ Qux (32 tiles), kv = -Inv @ q_u ----
    for (int tile = wave; tile < 32; tile += RIC_NW) {
      int r0 = (tile >> 3) * 16, c0 = (tile & 7) * 16;
      v8f acc = {};
      for (int k0 = 0; k0 < LQR_NU; k0 += 4) {
        v2f a = load_a_rm(Inv, LQR_NU, r0, k0, lane);
        v2f b = load_b_rm(Qux, LQR_NX, k0, c0, lane);
        acc = wmma_f32(a, b, acc);
      }
      store_c_rm_neg(Km, LQR_NX, r0, c0, acc, lane);
    }
    for (int i = tid; i < LQR_NU; i += RIC_NT) {
      float s = 0.0f;
      for (int u = 0; u < LQR_NU; ++u) s += Inv[i * LQR_NU + u] * qq[LQR_NX + u];
      kv[i] = -s;
    }
    __syncthreads();

    // ---- phase 5: V' = Qxx + Qux^T K (64 tiles), v' = q_x + K^T q_u,
    //               stream K,k to global ----
    for (int tile = wave; tile < 64; tile += RIC_NW) {
      int r0 = (tile >> 3) * 16, c0 = (tile & 7) * 16;
      v8f acc = load_c_rm(Qxx, LQR_NX, r0, c0, lane);
      for (int k0 = 0; k0 < LQR_NU; k0 += 4) {
        v2f a = load_a_tr(Qux, LQR_NX, r0, k0, lane);  // (Qux^T)[m,k]
        v2f b = load_b_rm(Km, LQR_NX, k0, c0, lane);
        acc = wmma_f32(a, b, acc);
      }
      store_c_rm(V, LQR_NX, r0, c0, acc, lane);
    }
    for (int i = tid; i < LQR_NX; i += RIC_NT) {
      float s = qq[i];
      for (int u = 0; u < LQR_NU; ++u) s += Km[u * LQR_NX + i] * qq[LQR_NX + u];
      vv[i] = s;
    }
    {
      float* Kd = Kout + (size_t)t * (LQR_NU * LQR_NX);
      for (int i = tid; i < LQR_NU * LQR_NX; i += RIC_NT) Kd[i] = Km[i];
      float* kd = kout + (size_t)t * LQR_NU;
      for (int i = tid; i < LQR_NU; i += RIC_NT) kd[i] = kv[i];
    }
    __syncthreads();
  }
}

// ---------------------------------------------------------------------------
// Kernel 2: forward rollout (sequential mat-vecs; latency-bound, plain VALU)
// ---------------------------------------------------------------------------
__global__ void __launch_bounds__(256)
lqr_rollout(const float* __restrict__ F, const float* __restrict__ fv,
            const float* __restrict__ C, const float* __restrict__ cv,
            const float* __restrict__ x0,
            const float* __restrict__ Kg, const float* __restrict__ kg,
            float* __restrict__ states, float* __restrict__ us,
            float* __restrict__ costs) {
  __shared__ float xs[LQR_NX], xn[LQR_NX], z[LQR_ND], red[256];
  const int tid = threadIdx.x;

  for (int i = tid; i < LQR_NX; i += 256) {
    xs[i] = x0[i];
    states[i] = x0[i];
  }
  __syncthreads();

  for (int s = 0; s < LQR_T; ++s) {
    const float* K  = Kg + (size_t)(LQR_T - 1 - s) * (LQR_NU * LQR_NX);
    const float* kk = kg + (size_t)(LQR_T - 1 - s) * LQR_NU;
    for (int i = tid; i < LQR_NX; i += 256) z[i] = xs[i];
    for (int a = tid; a < LQR_NU; a += 256) {
      float u = kk[a];
      for (int i = 0; i < LQR_NX; ++i) u += K[a * LQR_NX + i] * xs[i];
      z[LQR_NX + a] = u;
      us[(size_t)s * LQR_NU + a] = u;
    }
    __syncthreads();

    for (int i = tid; i < LQR_NX; i += 256) {
      float v = fv[i];
      for (int j = 0; j < LQR_ND; ++j) v += F[i * LQR_ND + j] * z[j];
      xn[i] = v;
    }
    float part = 0.0f;
    for (int j = tid; j < LQR_ND; j += 256) {
      float cz = 0.0f;
      for (int k = 0; k < LQR_ND; ++k) cz += C[j * LQR_ND + k] * z[k];
      part += z[j] * (0.5f * cz + cv[j]);
    }
    red[tid] = part;
    __syncthreads();
    for (int off = 128; off > 0; off >>= 1) {
      if (tid < off) red[tid] += red[tid + off];
      __syncthreads();
    }
    if (tid == 0) costs[s] = red[0];
    for (int i = tid; i < LQR_NX; i += 256) {
      xs[i] = xn[i];
      states[(size_t)(s + 1) * LQR_NX + i] = xn[i];
    }
    __syncthreads();
  }

  // final cost: 0.5 x^T Cxx x + x^T c_x
  float part = 0.0f;
  for (int j = tid; j < LQR_NX; j += 256) {
    float cz = 0.0f;
    for (int k = 0; k < LQR_NX; ++k) cz += C[j * LQR_ND + k] * xs[k];
    part += xs[j] * (0.5f * cz + cv[j]);
  }
  red[tid] = part;
  __syncthreads();
  for (int off = 128; off > 0; off >>= 1) {
    if (tid < off) red[tid] += red[tid + off];
    __syncthreads();
  }
  if (tid == 0) costs[LQR_T] = red[0];
}

// ---------------------------------------------------------------------------
extern "C" void kernel_launch(void* const* d_in, const int* in_sizes, int n_in,
                              void* d_out, int out_size, void* d_ws,
                              size_t ws_size, hipStream_t stream) {
  (void)in_sizes; (void)n_in; (void)out_size; (void)ws_size;
  const float* F  = (const float*)d_in[0];  // 128x192
  const float* f  = (const float*)d_in[1];  // 128
  const float* C  = (const float*)d_in[2];  // 192x192
  const float* c  = (const float*)d_in[3];  // 192
  const float* x0 = (const float*)d_in[4];  // 128
  // d_in[5] is T (=2048), compiled in as LQR_T.

  float* Kg = (float*)d_ws;                               // 2048*64*128 floats
  float* kg = Kg + (size_t)LQR_T * LQR_NU * LQR_NX;       // 2048*64 floats

  float* out    = (float*)d_out;
  float* states = out;                                    // (T+1)*128
  float* us     = out + (size_t)(LQR_T + 1) * LQR_NX;     // T*64
  float* costs  = us + (size_t)LQR_T * LQR_NU;            // T+1

  const size_t ldsBytes = 70208u * sizeof(float);         // 280832 B
  (void)hipFuncSetAttribute((const void*)lqr_riccati,
                            hipFuncAttributeMaxDynamicSharedMemorySize,
                            (int)ldsBytes);

  lqr_riccati<<<1, RIC_NT, ldsBytes, stream>>>(F, f, C, c, Kg, kg);
  lqr_rollout<<<1, 256, 0, stream>>>(F, f, C, c, x0, Kg, kg, states, us, costs);
}